// SpatialAttention_68530498175382
// MI455X (gfx1250) — compile-verified
//
#include <hip/hip_runtime.h>
#include <hip/hip_bf16.h>
#include <math.h>

// Problem constants (from reference setup_inputs)
constexpr int B  = 4;
constexpr int N  = 256;
constexpr int T  = 128;
constexpr int D  = 512;
constexpr int H  = 8;
constexpr int HD = 64;          // D / H
constexpr int BN = B * N;       // 1024

typedef __attribute__((ext_vector_type(2))) float v2f;
typedef __attribute__((ext_vector_type(4))) float v4f;
typedef __attribute__((ext_vector_type(8))) float v8f;

// ---------------------------------------------------------------------------
// Kernel 1: mean over T.  x (B,N,T,D) -> x_repr (B*N, D).
// Streaming read of 256MB: float4 non-temporal loads, coalesced across d.
// ---------------------------------------------------------------------------
__global__ void mean_kernel(const v4f* __restrict__ x4, v4f* __restrict__ xr4) {
  constexpr int D4 = D / 4;
  constexpr int total = BN * D4;
  int idx = blockIdx.x * blockDim.x + threadIdx.x;
  if (idx >= total) return;
  int bn = idx / D4;
  int d4 = idx - bn * D4;
  const v4f* p = x4 + (size_t)bn * T * D4 + d4;
  v4f s = {0.f, 0.f, 0.f, 0.f};
#pragma unroll 4
  for (int t = 0; t < T; ++t)
    s += __builtin_nontemporal_load(p + (size_t)t * D4);
  xr4[idx] = s * (1.0f / (float)T);
}

// ---------------------------------------------------------------------------
// Register-blocked WMMA f32 GEMM, C = alpha * (A @ W^T) + bias.
// Each wave computes a (16*MB) x (16*NB) macrotile: MB A-fragments and NB
// B-fragments feed MB*NB V_WMMA_F32_16X16X4_F32 per k-step (fragment reuse
// cuts L2 traffic by ~NB x for A and ~MB x for W vs unblocked).
// Batched via blockIdx.y with split offsets: off = (bz/nH)*s1 + (bz%nH)*s2.
// Fragment layout (ISA 7.12.2, 32-bit):
//   A: lane&15 = M row, lane>>4 selects K pair {2h, 2h+1} -> contiguous v2f
//   B: lane&15 = N col, same K striping -> row of W, contiguous v2f
//   C: VGPR r, lane<16 -> M=r ; lane>=16 -> M=r+8 ; N = lane&15
// ---------------------------------------------------------------------------
template <int MB, int NB>
__global__ void gemm_abt_wmma(const float* __restrict__ A, int lda, long sA1, long sA2,
                              const float* __restrict__ W, int ldw, long sW1, long sW2,
                              const float* __restrict__ bias,
                              float* __restrict__ C, int ldc, long sC1, long sC2,
                              int M, int Nn, int K, int nH, float alpha) {
  int wave = threadIdx.x >> 5;
  int lane = threadIdx.x & 31;
  int tile = blockIdx.x * (blockDim.x >> 5) + wave;
  int tilesN = Nn / (16 * NB);
  int tilesTotal = (M / (16 * MB)) * tilesN;
  if (tile >= tilesTotal) return;            // wave-uniform guard (EXEC stays all-1)

  int bz = blockIdx.y;
  const float* Ab = A + (long)(bz / nH) * sA1 + (long)(bz % nH) * sA2;
  const float* Wb = W + (long)(bz / nH) * sW1 + (long)(bz % nH) * sW2;
  float*       Cb = C + (long)(bz / nH) * sC1 + (long)(bz % nH) * sC2;

  int tm = (tile / tilesN) * (16 * MB);
  int tn = (tile % tilesN) * (16 * NB);
  int lo = lane & 15;
  int hi = lane >> 4;

  const float* arow[MB];
  const float* wrow[NB];
#pragma unroll
  for (int i = 0; i < MB; ++i)
    arow[i] = Ab + (size_t)(tm + 16 * i + lo) * lda + 2 * hi;
#pragma unroll
  for (int j = 0; j < NB; ++j)
    wrow[j] = Wb + (size_t)(tn + 16 * j + lo) * ldw + 2 * hi;

  v8f acc[MB][NB] = {};
  for (int k0 = 0; k0 < K; k0 += 4) {
    v2f a[MB], b[NB];
#pragma unroll
    for (int i = 0; i < MB; ++i) a[i] = *(const v2f*)(arow[i] + k0);
#pragma unroll
    for (int j = 0; j < NB; ++j) b[j] = *(const v2f*)(wrow[j] + k0);
#pragma unroll
    for (int i = 0; i < MB; ++i)
#pragma unroll
      for (int j = 0; j < NB; ++j)
        acc[i][j] = __builtin_amdgcn_wmma_f32_16x16x4_f32(
            false, a[i], false, b[j], (short)0, acc[i][j], false, false);
  }

#pragma unroll
  for (int j = 0; j < NB; ++j) {
    int cn = tn + 16 * j + lo;
    float bv = bias ? bias[cn] : 0.0f;
#pragma unroll
    for (int i = 0; i < MB; ++i)
#pragma unroll
      for (int r = 0; r < 8; ++r) {
        int cm = tm + 16 * i + r + 8 * hi;
        Cb[(size_t)cm * ldc + cn] = acc[i][j][r] * alpha + bv;
      }
  }
}

// ---------------------------------------------------------------------------
// Register-blocked WMMA f32 GEMM, C = A @ Bm (Bm row-major K x Nn).
// Used for O = w @ V.  B-fragment rows are strided (ldb apart) -> scalar loads.
// ---------------------------------------------------------------------------
template <int MB, int NB>
__global__ void gemm_ab_wmma(const float* __restrict__ A, int lda, long sA1, long sA2,
                             const float* __restrict__ Bm, int ldb, long sB1, long sB2,
                             float* __restrict__ C, int ldc, long sC1, long sC2,
                             int M, int Nn, int K, int nH) {
  int wave = threadIdx.x >> 5;
  int lane = threadIdx.x & 31;
  int tile = blockIdx.x * (blockDim.x >> 5) + wave;
  int tilesN = Nn / (16 * NB);
  int tilesTotal = (M / (16 * MB)) * tilesN;
  if (tile >= tilesTotal) return;

  int bz = blockIdx.y;
  const float* Ab = A  + (long)(bz / nH) * sA1 + (long)(bz % nH) * sA2;
  const float* Bb = Bm + (long)(bz / nH) * sB1 + (long)(bz % nH) * sB2;
  float*       Cb = C  + (long)(bz / nH) * sC1 + (long)(bz % nH) * sC2;

  int tm = (tile / tilesN) * (16 * MB);
  int tn = (tile % tilesN) * (16 * NB);
  int lo = lane & 15;
  int hi = lane >> 4;

  const float* arow[MB];
#pragma unroll
  for (int i = 0; i < MB; ++i)
    arow[i] = Ab + (size_t)(tm + 16 * i + lo) * lda + 2 * hi;

  v8f acc[MB][NB] = {};
  for (int k0 = 0; k0 < K; k0 += 4) {
    v2f a[MB], b[NB];
#pragma unroll
    for (int i = 0; i < MB; ++i) a[i] = *(const v2f*)(arow[i] + k0);
#pragma unroll
    for (int j = 0; j < NB; ++j) {
      int cn = tn + 16 * j + lo;
      b[j].x = Bb[(size_t)(k0 + 2 * hi)     * ldb + cn];
      b[j].y = Bb[(size_t)(k0 + 2 * hi + 1) * ldb + cn];
    }
#pragma unroll
    for (int i = 0; i < MB; ++i)
#pragma unroll
      for (int j = 0; j < NB; ++j)
        acc[i][j] = __builtin_amdgcn_wmma_f32_16x16x4_f32(
            false, a[i], false, b[j], (short)0, acc[i][j], false, false);
  }

#pragma unroll
  for (int j = 0; j < NB; ++j) {
    int cn = tn + 16 * j + lo;
#pragma unroll
    for (int i = 0; i < MB; ++i)
#pragma unroll
      for (int r = 0; r < 8; ++r) {
        int cm = tm + 16 * i + r + 8 * hi;
        Cb[(size_t)cm * ldc + cn] = acc[i][j][r];
      }
  }
}

// ---------------------------------------------------------------------------
// Kernel: fused haversine-distance bias + mask + softmax, in place on scores.
// One block (256 threads) per (b,h,n) row; thread m handles one column.
// ---------------------------------------------------------------------------
__global__ void softmax_bias_kernel(float* __restrict__ S,              // (B,H,N,N)
                                    const float* __restrict__ coords,   // (B,N,2)
                                    const int*   __restrict__ mask,     // (B,N)
                                    const float* __restrict__ Wd1,      // (HD,1)
                                    const float* __restrict__ bd1,      // (HD)
                                    const float* __restrict__ Wd2,      // (H,HD)
                                    const float* __restrict__ bd2) {    // (H)
  constexpr float DEG2RAD = 0.017453292519943295f;
  constexpr float NEG_HUGE = -3.0e38f;

  int row = blockIdx.x;                 // b*H*N + h*N + n
  int n = row % N;
  int h = (row / N) % H;
  int b = row / (N * H);
  int m = threadIdx.x;

  // haversine(b, n, m)
  float lat1 = coords[((size_t)(b * N + n)) * 2 + 0] * DEG2RAD;
  float lon1 = coords[((size_t)(b * N + n)) * 2 + 1] * DEG2RAD;
  float lat2 = coords[((size_t)(b * N + m)) * 2 + 0] * DEG2RAD;
  float lon2 = coords[((size_t)(b * N + m)) * 2 + 1] * DEG2RAD;
  float sdlat = sinf((lat2 - lat1) * 0.5f);
  float sdlon = sinf((lon2 - lon1) * 0.5f);
  float a = sdlat * sdlat + cosf(lat1) * cosf(lat2) * sdlon * sdlon;
  a = fminf(fmaxf(a, 0.0f), 1.0f);
  float dist = 6371.0f * 2.0f * atan2f(sqrtf(a), sqrtf(1.0f - a));

  // bias MLP: relu(dist*Wd1 + bd1) dotted with Wd2[h]
  float bias = bd2[h];
#pragma unroll 8
  for (int j = 0; j < HD; ++j) {
    float hm = fmaxf(dist * Wd1[j] + bd1[j], 0.0f);
    bias += Wd2[h * HD + j] * hm;
  }

  bool masked = (mask[b * N + m] == 0);
  size_t sidx = (size_t)row * N + m;
  float s = masked ? NEG_HUGE : (S[sidx] + bias);

  __shared__ float red[N];
  red[m] = s;
  __syncthreads();
#pragma unroll
  for (int off = N / 2; off > 0; off >>= 1) {
    if (m < off) red[m] = fmaxf(red[m], red[m + off]);
    __syncthreads();
  }
  float mx = red[0];
  __syncthreads();

  float e = masked ? 0.0f : expf(s - mx);
  red[m] = e;
  __syncthreads();
#pragma unroll
  for (int off = N / 2; off > 0; off >>= 1) {
    if (m < off) red[m] += red[m + off];
    __syncthreads();
  }
  float sum = red[0];

  S[sidx] = e / sum;
}

// ---------------------------------------------------------------------------
// Kernel: broadcast (B,N,D) -> (B,N,T,D).  Streaming 256MB write with
// non-temporal b128 stores; source rows (2MB) stay resident in L2.
// ---------------------------------------------------------------------------
__global__ void broadcast_kernel(const v4f* __restrict__ r4, v4f* __restrict__ out4) {
  constexpr int D4 = D / 4;
  constexpr size_t total = (size_t)BN * T * D4;
  size_t i = (size_t)blockIdx.x * blockDim.x + threadIdx.x;
  if (i >= total) return;
  size_t bn = i / ((size_t)T * D4);
  int d4 = (int)(i % D4);
  v4f v = r4[bn * D4 + d4];
  __builtin_nontemporal_store(v, out4 + i);
}

// ---------------------------------------------------------------------------
// Host launcher
// ---------------------------------------------------------------------------
extern "C" void kernel_launch(void* const* d_in, const int* in_sizes, int n_in,
                              void* d_out, int out_size, void* d_ws, size_t ws_size,
                              hipStream_t stream) {
  const float* x      = (const float*)d_in[0];
  const float* coords = (const float*)d_in[1];
  const int*   smask  = (const int*)  d_in[2];
  const float* Wq = (const float*)d_in[3];  const float* bq = (const float*)d_in[4];
  const float* Wk = (const float*)d_in[5];  const float* bk = (const float*)d_in[6];
  const float* Wv = (const float*)d_in[7];  const float* bv = (const float*)d_in[8];
  const float* Wo = (const float*)d_in[9];  const float* bo = (const float*)d_in[10];
  const float* Wd1 = (const float*)d_in[11]; const float* bd1 = (const float*)d_in[12];
  const float* Wd2 = (const float*)d_in[13]; const float* bd2 = (const float*)d_in[14];
  float* out = (float*)d_out;

  // Workspace layout (floats). BN*D = 524288 per (B,N,D) buffer; scores = B*H*N*N.
  float* ws  = (float*)d_ws;
  float* xr  = ws;                    // (B,N,D)  x_repr
  float* Q   = ws + 1 * 524288;       // (B,N,D)
  float* Kt  = ws + 2 * 524288;       // (B,N,D)
  float* V   = ws + 3 * 524288;       // (B,N,D)
  float* S   = ws + 4 * 524288;       // (B,H,N,N) scores -> softmax weights (8MB)
  float* O   = xr;                    // reuse: attention output (B,N,D)
  float* R   = Q;                     // reuse: final projected output (B,N,D)

  constexpr int MB = 2, NB = 4;       // 32 x 64 macrotile per wave
  constexpr int WPB = 8;              // waves per 256-thread block

  // 1) mean over T
  {
    constexpr int total = BN * (D / 4);
    mean_kernel<<<(total + 255) / 256, 256, 0, stream>>>((const v4f*)x, (v4f*)xr);
  }

  // 2) Q/K/V projections: M=1024, Nn=512, K=512 -> (1024/32)*(512/64)=256 macrotiles
  {
    int tiles = (BN / (16 * MB)) * (D / (16 * NB));   // 256
    int blocks = (tiles + WPB - 1) / WPB;             // 32
    gemm_abt_wmma<MB, NB><<<dim3(blocks, 1), 256, 0, stream>>>(
        xr, D, 0, 0, Wq, D, 0, 0, bq, Q,  D, 0, 0, BN, D, D, 1, 1.0f);
    gemm_abt_wmma<MB, NB><<<dim3(blocks, 1), 256, 0, stream>>>(
        xr, D, 0, 0, Wk, D, 0, 0, bk, Kt, D, 0, 0, BN, D, D, 1, 1.0f);
    gemm_abt_wmma<MB, NB><<<dim3(blocks, 1), 256, 0, stream>>>(
        xr, D, 0, 0, Wv, D, 0, 0, bv, V,  D, 0, 0, BN, D, D, 1, 1.0f);
  }

  // 3) scores[b,h] = (Q_bh @ K_bh^T) / sqrt(hd): M=N=256, K=64, 32 batches (b,h)
  //    per-batch macrotiles = (256/32)*(256/64) = 32 -> 4 blocks
  gemm_abt_wmma<MB, NB><<<dim3(4, B * H), 256, 0, stream>>>(
      Q,  D, (long)N * D, HD,
      Kt, D, (long)N * D, HD,
      nullptr,
      S,  N, (long)H * N * N, (long)N * N,
      N, N, HD, H, 0.125f);

  // 4) fused haversine bias + mask + softmax (in place on S)
  softmax_bias_kernel<<<B * H * N, N, 0, stream>>>(S, coords, smask,
                                                   Wd1, bd1, Wd2, bd2);

  // 5) O[b,h] = w_bh @ V_bh : M=256, Nn=64, K=256, 32 batches
  //    per-batch macrotiles = (256/32)*(64/64) = 8 -> 1 block
  gemm_ab_wmma<MB, NB><<<dim3(1, B * H), 256, 0, stream>>>(
      S, N, (long)H * N * N, (long)N * N,
      V, D, (long)N * D, HD,
      O, D, (long)N * D, HD,
      N, HD, N, H);

  // 6) output projection: R = O @ Wo^T + bo
  {
    int tiles = (BN / (16 * MB)) * (D / (16 * NB));
    int blocks = (tiles + WPB - 1) / WPB;
    gemm_abt_wmma<MB, NB><<<dim3(blocks, 1), 256, 0, stream>>>(
        O, D, 0, 0, Wo, D, 0, 0, bo, R, D, 0, 0, BN, D, D, 1, 1.0f);
  }

  // 7) broadcast over T (streaming NT stores)
  {
    constexpr size_t total = (size_t)BN * T * (D / 4);
    broadcast_kernel<<<(unsigned)((total + 255) / 256), 256, 0, stream>>>(
        (const v4f*)R, (v4f*)out);
  }
}